// DBLoss_80848464380152
// MI455X (gfx1250) — compile-verified
//
#include <hip/hip_runtime.h>
#include <hip/hip_bf16.h>
#include <math.h>

#define NBATCH 16
#define HW     409600          // 640*640
#define EPSF   1e-7f

// ---- workspace layout (bytes) ----
#define WS_ACC     0                        // 5 doubles: cm, ss, sb, ctm, sl1
#define WS_POS     64                       // 16 u32 pos counts
#define WS_RANK    128                      // 16 u32 residual rank
#define WS_PREFIX  192                      // 16 u32 bit prefix
#define WS_THRESH  256                      // 16 f32 thresholds
#define WS_VALID   320                      // 16 u32 valid flags
#define WS_H1      512                      // 16*2048 u32
#define WS_H2      (WS_H1 + 16*2048*4)      // 16*2048 u32
#define WS_H3      (WS_H2 + 16*2048*4)      // 16*1024 u32
#define WS_TOTAL   (WS_H3 + 16*1024*4)      // = 328192 bytes

// ---- CDNA5 async global->LDS path (guarded; falls back to float4 loads) ----
#if __has_builtin(__builtin_amdgcn_global_load_async_to_lds_b128)
#define HAS_ASYNC_LDS 1
typedef int v4i __attribute__((vector_size(16)));
typedef __attribute__((address_space(1))) v4i* g_v4i_p;   // global int4*
typedef __attribute__((address_space(3))) v4i* l_v4i_p;   // LDS int4*
__device__ __forceinline__ void async_copy16(const float* gsrc, float* lds_dst){
  __builtin_amdgcn_global_load_async_to_lds_b128(
      (g_v4i_p)gsrc, (l_v4i_p)lds_dst, /*offset=*/0, /*cpol=*/0);
}
#else
#define HAS_ASYNC_LDS 0
#endif

#if __has_builtin(__builtin_amdgcn_s_wait_asynccnt)
#define WAIT_ASYNC(n) __builtin_amdgcn_s_wait_asynccnt(n)
#else
#define WAIT_ASYNC(n) asm volatile("s_wait_asynccnt %0" :: "i"(n) : "memory")
#endif

// order-preserving float<->uint mapping (total order, ascending)
__device__ __forceinline__ unsigned ordf(float x){
  unsigned b = __float_as_uint(x);
  return (b & 0x80000000u) ? ~b : (b | 0x80000000u);
}
__device__ __forceinline__ float unordf(unsigned u){
  unsigned b = (u & 0x80000000u) ? (u & 0x7fffffffu) : ~u;
  return __uint_as_float(b);
}

__global__ void ws_zero(uint4* __restrict__ ws, int nvec){
  int i = blockIdx.x * blockDim.x + threadIdx.x;
  if (i < nvec) ws[i] = make_uint4(0u, 0u, 0u, 0u);
}

// Radix-select histogram pass. stage0: bins = u>>21 (+ pos count).
// stage1: restricted to (u>>21)==prefix, bins = (u>>10)&0x7ff.
// stage2: restricted to (u>>10)==prefix, bins = u&0x3ff.
__global__ void hist_pass(const float* __restrict__ outputs,
                          const float* __restrict__ gts,
                          unsigned* __restrict__ hist,
                          unsigned* __restrict__ posArr,
                          const unsigned* __restrict__ prefix,
                          int stage){
  __shared__ unsigned lh[2048];
  __shared__ unsigned lpos;
  const int tid = threadIdx.x;
  const int b   = blockIdx.y;
  const int nb  = (stage == 2) ? 1024 : 2048;

  const size_t sbase = (size_t)b * 3u * HW;   // shrink map = plane 0
  const size_t gbase = (size_t)b * HW;
  const int i0 = (blockIdx.x * 256 + tid) * 4;

  float sv[4], gv[4];
#if HAS_ASYNC_LDS
  __shared__ float stage_s[1024];
  __shared__ float stage_g[1024];
  // kick off DMA into LDS, overlap with histogram init
  async_copy16(outputs + sbase + i0, &stage_s[tid * 4]);
  async_copy16(gts + gbase + i0,     &stage_g[tid * 4]);
  for (int i = tid; i < nb; i += 256) lh[i] = 0u;
  if (tid == 0) lpos = 0u;
  WAIT_ASYNC(0);
  __syncthreads();
  #pragma unroll
  for (int j = 0; j < 4; ++j){ sv[j] = stage_s[tid*4+j]; gv[j] = stage_g[tid*4+j]; }
#else
  for (int i = tid; i < nb; i += 256) lh[i] = 0u;
  if (tid == 0) lpos = 0u;
  __syncthreads();
  const float4 s4 = *(const float4*)(outputs + sbase + i0);
  const float4 g4 = *(const float4*)(gts + gbase + i0);
  sv[0]=s4.x; sv[1]=s4.y; sv[2]=s4.z; sv[3]=s4.w;
  gv[0]=g4.x; gv[1]=g4.y; gv[2]=g4.z; gv[3]=g4.w;
#endif

  const unsigned pfx = (stage == 0) ? 0u : prefix[b];
  unsigned pc = 0;
  #pragma unroll
  for (int j = 0; j < 4; ++j){
    const bool pos = gv[j] > 0.5f;
    if (stage == 0){
      if (pos) pc++;
      else     atomicAdd(&lh[ordf(sv[j]) >> 21], 1u);
    } else if (stage == 1){
      if (!pos){ unsigned u = ordf(sv[j]); if ((u >> 21) == pfx) atomicAdd(&lh[(u >> 10) & 0x7ffu], 1u); }
    } else {
      if (!pos){ unsigned u = ordf(sv[j]); if ((u >> 10) == pfx) atomicAdd(&lh[u & 0x3ffu], 1u); }
    }
  }
  if (stage == 0 && pc) atomicAdd(&lpos, pc);
  __syncthreads();

  unsigned* gh = hist + (size_t)b * nb;
  for (int i = tid; i < nb; i += 256){
    unsigned c = lh[i];
    if (c) atomicAdd(&gh[i], c);
  }
  if (stage == 0 && tid == 0 && lpos) atomicAdd(&posArr[b], lpos);
}

// One thread per batch: scan histogram from the top to locate the bin that
// contains the k-th largest negative score; update prefix / residual rank.
__global__ void select_pass(const unsigned* __restrict__ hist,
                            const unsigned* __restrict__ posArr,
                            unsigned* __restrict__ rank,
                            unsigned* __restrict__ prefix,
                            float* __restrict__ thresh,
                            unsigned* __restrict__ validArr,
                            int stage){
  const int b = threadIdx.x;
  if (b >= NBATCH) return;
  const int nb = (stage == 2) ? 1024 : 2048;
  unsigned k;
  if (stage == 0){
    const unsigned pos    = posArr[b];
    const unsigned negTot = (unsigned)HW - pos;
    unsigned kk = pos * 3u; if (kk > negTot) kk = negTot;
    const bool valid = (pos > 0u) && (kk > 0u);
    validArr[b] = valid ? 1u : 0u;
    if (!valid){ rank[b] = 0u; prefix[b] = 0xffffffffu; thresh[b] = INFINITY; return; }
    k = kk;
  } else {
    k = rank[b];
    if (k == 0u) return;
  }
  const unsigned* h = hist + (size_t)b * nb;
  unsigned cum = 0; int sel = 0;
  for (int bin = nb - 1; bin >= 0; --bin){
    const unsigned c = h[bin];
    if (cum + c >= k){ sel = bin; break; }
    cum += c;
  }
  if (stage == 0)      { prefix[b] = (unsigned)sel;                         rank[b] = k - cum; }
  else if (stage == 1) { prefix[b] = (prefix[b] << 11) | (unsigned)sel;     rank[b] = k - cum; }
  else                 { thresh[b] = unordf((prefix[b] << 10) | (unsigned)sel); }
}

// Fused masked-BCE (shrink + binary) and masked-L1 (threshold) reduction.
// Double-buffered async global->LDS staging of all 5 planes when available.
__global__ void loss_pass(const float* __restrict__ outputs,
                          const float* __restrict__ gts,
                          const float* __restrict__ gtt,
                          const float* __restrict__ thresh,
                          const unsigned* __restrict__ validArr,
                          double* __restrict__ acc){
  const int tid = threadIdx.x;
  const int b   = blockIdx.y;
  const float th  = thresh[b];
  const bool  val = validArr[b] != 0u;
  const size_t b3 = (size_t)b * 3u * HW;
  const size_t b1 = (size_t)b * HW;

  float cm = 0.f, ss = 0.f, sb = 0.f, ctm = 0.f, sl1 = 0.f;

#if HAS_ASYNC_LDS
  __shared__ float stg[2][5][1024];
  auto issue = [&](int buf, int chunk){
    const int io = (chunk * 256 + tid) * 4;
    async_copy16(outputs + b3 + io,                     &stg[buf][0][tid * 4]);
    async_copy16(outputs + b3 + HW + io,                &stg[buf][1][tid * 4]);
    async_copy16(outputs + b3 + 2u*(size_t)HW + io,     &stg[buf][2][tid * 4]);
    async_copy16(gts + b1 + io,                         &stg[buf][3][tid * 4]);
    async_copy16(gtt + b1 + io,                         &stg[buf][4][tid * 4]);
  };
  issue(0, blockIdx.x);                 // buffer 0
  issue(1, blockIdx.x + gridDim.x);     // buffer 1 (in flight behind buffer 0)
#endif

  #pragma unroll 1
  for (int it = 0; it < 2; ++it){
    float sv[4], tv[4], bv[4], gv[4], hv[4];
#if HAS_ASYNC_LDS
    if (it == 0) { WAIT_ASYNC(5); }     // async ops complete in order: first 5 done
    else         { WAIT_ASYNC(0); }
    __syncthreads();
    #pragma unroll
    for (int j = 0; j < 4; ++j){
      sv[j] = stg[it][0][tid*4+j];
      tv[j] = stg[it][1][tid*4+j];
      bv[j] = stg[it][2][tid*4+j];
      gv[j] = stg[it][3][tid*4+j];
      hv[j] = stg[it][4][tid*4+j];
    }
#else
    const int i0 = ((blockIdx.x + it * gridDim.x) * 256 + tid) * 4;
    if (it == 0){
      const int i1 = ((blockIdx.x + gridDim.x) * 256 + tid) * 4;
      __builtin_prefetch(outputs + b3 + i1, 0, 0);                 // global_prefetch_b8
      __builtin_prefetch(outputs + b3 + HW + i1, 0, 0);
      __builtin_prefetch(outputs + b3 + 2u*(size_t)HW + i1, 0, 0);
      __builtin_prefetch(gts + b1 + i1, 0, 0);
      __builtin_prefetch(gtt + b1 + i1, 0, 0);
    }
    const float4 s4 = *(const float4*)(outputs + b3 + i0);
    const float4 t4 = *(const float4*)(outputs + b3 + HW + i0);
    const float4 m4 = *(const float4*)(outputs + b3 + 2u*(size_t)HW + i0);
    const float4 g4 = *(const float4*)(gts + b1 + i0);
    const float4 h4 = *(const float4*)(gtt + b1 + i0);
    sv[0]=s4.x; sv[1]=s4.y; sv[2]=s4.z; sv[3]=s4.w;
    tv[0]=t4.x; tv[1]=t4.y; tv[2]=t4.z; tv[3]=t4.w;
    bv[0]=m4.x; bv[1]=m4.y; bv[2]=m4.z; bv[3]=m4.w;
    gv[0]=g4.x; gv[1]=g4.y; gv[2]=g4.z; gv[3]=g4.w;
    hv[0]=h4.x; hv[1]=h4.y; hv[2]=h4.z; hv[3]=h4.w;
#endif

    #pragma unroll
    for (int j = 0; j < 4; ++j){
      const bool pos = gv[j] > 0.5f;
      const bool m   = val && (pos || (sv[j] >= th));
      if (m){
        cm += 1.0f;
        float p = 1.0f / (1.0f + expf(-sv[j]));
        p = fminf(fmaxf(p, EPSF), 1.0f - EPSF);
        ss += pos ? -logf(p) : -logf(1.0f - p);
        float pb = 1.0f / (1.0f + expf(-bv[j]));
        pb = fminf(fmaxf(pb, EPSF), 1.0f - EPSF);
        sb += pos ? -logf(pb) : -logf(1.0f - pb);
      }
      if ((hv[j] > 0.0f) || (gv[j] > 0.0f)){
        ctm += 1.0f;
        const float pt = 1.0f / (1.0f + expf(-tv[j]));
        sl1 += fabsf(pt - hv[j]);
      }
    }
  }

  __shared__ float sd[5][256];
  sd[0][tid]=cm; sd[1][tid]=ss; sd[2][tid]=sb; sd[3][tid]=ctm; sd[4][tid]=sl1;
  __syncthreads();
  for (int s = 128; s > 0; s >>= 1){
    if (tid < s){
      #pragma unroll
      for (int q = 0; q < 5; ++q) sd[q][tid] += sd[q][tid + s];
    }
    __syncthreads();
  }
  if (tid < 5) atomicAdd(&acc[tid], (double)sd[tid][0]);   // global_atomic_add_f64
}

__global__ void finalize_k(const double* __restrict__ acc, float* __restrict__ out){
  const double cm = acc[0], ss = acc[1], sb = acc[2], ctm = acc[3], sl1 = acc[4];
  const float ls = (cm  > 0.0) ? (float)(ss  / fmax(cm , 1.0)) : 0.0f;
  const float lb = (cm  > 0.0) ? (float)(sb  / fmax(cm , 1.0)) : 0.0f;
  const float lt = (ctm > 0.0) ? (float)(sl1 / fmax(ctm, 1.0)) : 0.0f;
  out[0] = ls + 1.0f * lb + 10.0f * lt;   // ALPHA=1, BETA=10
  out[1] = ls;
  out[2] = lb;
  out[3] = lt;
}

extern "C" void kernel_launch(void* const* d_in, const int* in_sizes, int n_in,
                              void* d_out, int out_size, void* d_ws, size_t ws_size,
                              hipStream_t stream){
  const float* outputs = (const float*)d_in[0];   // (16,3,640,640)
  const float* gts     = (const float*)d_in[1];   // (16,640,640)
  const float* gtt     = (const float*)d_in[2];   // (16,640,640)
  float* out = (float*)d_out;                     // 4 scalars

  char* ws = (char*)d_ws;
  double*   acc    = (double*)  (ws + WS_ACC);
  unsigned* posArr = (unsigned*)(ws + WS_POS);
  unsigned* rank   = (unsigned*)(ws + WS_RANK);
  unsigned* prefix = (unsigned*)(ws + WS_PREFIX);
  float*    thresh = (float*)   (ws + WS_THRESH);
  unsigned* validA = (unsigned*)(ws + WS_VALID);
  unsigned* h1     = (unsigned*)(ws + WS_H1);
  unsigned* h2     = (unsigned*)(ws + WS_H2);
  unsigned* h3     = (unsigned*)(ws + WS_H3);

  const int nvec = WS_TOTAL / 16;
  ws_zero<<<(nvec + 255) / 256, 256, 0, stream>>>((uint4*)ws, nvec);

  const dim3 gridH(HW / 1024, NBATCH);   // 256 thr * 4 elems = 1024/block, 400 blocks exact
  hist_pass<<<gridH, 256, 0, stream>>>(outputs, gts, h1, posArr, prefix, 0);
  select_pass<<<1, 32, 0, stream>>>(h1, posArr, rank, prefix, thresh, validA, 0);
  hist_pass<<<gridH, 256, 0, stream>>>(outputs, gts, h2, posArr, prefix, 1);
  select_pass<<<1, 32, 0, stream>>>(h2, posArr, rank, prefix, thresh, validA, 1);
  hist_pass<<<gridH, 256, 0, stream>>>(outputs, gts, h3, posArr, prefix, 2);
  select_pass<<<1, 32, 0, stream>>>(h3, posArr, rank, prefix, thresh, validA, 2);

  const dim3 gridL(HW / 2048, NBATCH);   // 2 iters of 1024 elems/block
  loss_pass<<<gridL, 256, 0, stream>>>(outputs, gts, gtt, thresh, validA, acc);
  finalize_k<<<1, 1, 0, stream>>>(acc, out);
}